// GroupFeature_532575945287
// MI455X (gfx1250) — compile-verified
//
#include <hip/hip_runtime.h>

typedef __attribute__((ext_vector_type(2))) float v2f;
typedef __attribute__((ext_vector_type(4))) float v4f;
typedef __attribute__((ext_vector_type(8))) float v8f;

#define B_    8
#define N_    2048
#define KNN   16
#define NTILE (N_ / 16)

// 128 threads = 4 waves; each wave owns one 16-query tile. The workgroup
// stages all 2048 candidate coords (+ negated squared norms) in LDS once.
// Each wave computes neg squared distances via v_wmma_f32_16x16x4_f32
// (A = candidates 16x4 with K3=1, B = queries 4x16 pre-scaled by 2 with
// K3 = -|q|^2, C = -|c|^2), so D = 2*c.q - |q|^2 - |c|^2 in one WMMA/tile.
// Per-lane branchless sorted top-16, then lane-pair merge per query.
__global__ __launch_bounds__(128) void GroupFeature_knn_wmma(
    const float* __restrict__ xyz,      // [B][N][3]
    float*       __restrict__ out_nbhd, // [B][N][KNN][3]
    int*         __restrict__ idx_out)  // [B][N][KNN]
{
    __shared__ float s_x[N_], s_y[N_], s_z[N_];
    __shared__ float s_nsq[N_];                 // -(x^2+y^2+z^2)
    __shared__ float s_mv[4][16][32];
    __shared__ int   s_mi[4][16][32];

    const int tid  = threadIdx.x;
    const int wave = tid >> 5;
    const int lane = tid & 31;
    const int b    = blockIdx.y;
    const int qt   = blockIdx.x * 4 + wave;     // query tile index
    const float* base = xyz + (size_t)b * N_ * 3;

    // Stage candidates once per workgroup (uniform loop, 16 iters/thread).
    for (int i = tid; i < N_; i += 128) {
        float x = base[i * 3 + 0];
        float y = base[i * 3 + 1];
        float z = base[i * 3 + 2];
        s_x[i] = x; s_y[i] = y; s_z[i] = z;
        s_nsq[i] = -(x * x + y * y + z * z);
    }
    __syncthreads();

    // B operand (queries on N dim), pre-scaled by 2; K=3 row = -|q|^2.
    const int qn = qt * 16 + (lane & 15);
    v2f bmat;
    if (lane < 16) { bmat.x = 2.0f * s_x[qn]; bmat.y = 2.0f * s_y[qn]; }
    else           { bmat.x = 2.0f * s_z[qn]; bmat.y = s_nsq[qn];      }

    const int hi8 = (lane < 16) ? 0 : 8;

    float lv[KNN];
    int   li[KNN];
#pragma unroll
    for (int j = 0; j < KNN; ++j) { lv[j] = -__builtin_inff(); li[j] = 0; }

    for (int t = 0; t < NTILE; ++t) {
        // A operand (candidates on M dim): 16x4 f32, K3 column = 1.0
        const int cm = t * 16 + (lane & 15);
        v2f amat;
        if (lane < 16) { amat.x = s_x[cm]; amat.y = s_y[cm]; }
        else           { amat.x = s_z[cm]; amat.y = 1.0f;    }

        // C operand: -|c_M|^2 ; this lane holds M = r + hi8 for vgpr r
        v8f cmat;
#pragma unroll
        for (int r = 0; r < 8; ++r) cmat[r] = s_nsq[t * 16 + hi8 + r];

        // D[M][N] = 2*c.q - |q|^2 - |c|^2  (neg squared distance)
        v8f d = __builtin_amdgcn_wmma_f32_16x16x4_f32(
            false, amat, false, bmat, (short)0, cmat, false, false);

        // Branchless sorted-insert (stream is index-ascending so strict '>'
        // reproduces jax.lax.top_k tie ordering: equal values keep low index).
#pragma unroll
        for (int r = 0; r < 8; ++r) {
            const float v  = d[r];
            const int   ci = t * 16 + hi8 + r;
            if (v > lv[KNN - 1]) {              // only divergent branch
                bool g[KNN];
#pragma unroll
                for (int s = 0; s < KNN; ++s) g[s] = (v > lv[s]);
#pragma unroll
                for (int s = KNN - 1; s >= 1; --s) {
                    lv[s] = g[s - 1] ? lv[s - 1] : (g[s] ? v  : lv[s]);
                    li[s] = g[s - 1] ? li[s - 1] : (g[s] ? ci : li[s]);
                }
                lv[0] = g[0] ? v  : lv[0];
                li[0] = g[0] ? ci : li[0];
            }
        }
    }

    // Lane l and l+16 both cover query l%16: merge their sorted 16-lists.
#pragma unroll
    for (int j = 0; j < KNN; ++j) {
        s_mv[wave][lane & 15][hi8 ? (16 + j) : j] = lv[j];
        s_mi[wave][lane & 15][hi8 ? (16 + j) : j] = li[j];
    }
    __syncthreads();

    if (lane < 16) {
        const int    q  = qt * 16 + lane;
        const float  qx = s_x[q], qy = s_y[q], qz = s_z[q];
        const size_t row = ((size_t)b * N_ + q) * KNN;
        int pa = 0, pb = 0;
        for (int j = 0; j < KNN; ++j) {
            float va = s_mv[wave][lane][pa], vb = s_mv[wave][lane][16 + pb];
            int   ia = s_mi[wave][lane][pa], ib = s_mi[wave][lane][16 + pb];
            bool  ta = (va > vb) || (va == vb && ia < ib);
            int  sel = ta ? ia : ib;
            if (ta) ++pa; else ++pb;
            idx_out[row + j]            = sel;
            out_nbhd[(row + j) * 3 + 0] = s_x[sel] - qx;
            out_nbhd[(row + j) * 3 + 1] = s_y[sel] - qy;
            out_nbhd[(row + j) * 3 + 2] = s_z[sel] - qz;
        }
    }
}

// Memory-bound feature gather: one 128-bit vector per thread, coalesced B128
// loads, non-temporal B128 stores (output is write-once; keep feat in L2).
__global__ __launch_bounds__(256) void GroupFeature_gather_feat(
    const v4f* __restrict__ feat,     // [B][N][64] x 128-bit
    const int* __restrict__ idx,      // [B*N*KNN]
    v4f*       __restrict__ out_feat) // [B*N*KNN][64] x 128-bit
{
    const int t   = blockIdx.x * 256 + threadIdx.x;
    const int row = t >> 6;          // (b*N + n)*KNN + k
    const int f4  = t & 63;
    const int b   = row >> 15;       // row / (N_*KNN) = row / 32768
    const int src = idx[row];
    const v4f val = feat[((size_t)b * N_ + src) * 64 + f4];
    __builtin_nontemporal_store(val, &out_feat[(size_t)row * 64 + f4]);
}

extern "C" void kernel_launch(void* const* d_in, const int* in_sizes, int n_in,
                              void* d_out, int out_size, void* d_ws, size_t ws_size,
                              hipStream_t stream) {
    const float* xyz  = (const float*)d_in[0];
    const float* feat = (const float*)d_in[1];

    float* out      = (float*)d_out;
    float* out_nbhd = out;                                   // [8,2048,16,3]
    float* out_feat = out + (size_t)B_ * N_ * KNN * 3;       // [8,2048,16,256]
    int*   idx      = (int*)d_ws;                            // 8*2048*16 ints = 1 MB

    GroupFeature_knn_wmma<<<dim3(N_ / 64, B_), 128, 0, stream>>>(xyz, out_nbhd, idx);

    const int total_f4 = B_ * N_ * KNN * 64;                 // 16,777,216
    GroupFeature_gather_feat<<<total_f4 / 256, 256, 0, stream>>>(
        (const v4f*)feat, idx, (v4f*)out_feat);
}